// ACTLossHead_46789373722673
// MI455X (gfx1250) — compile-verified
//
#include <hip/hip_runtime.h>
#include <hip/hip_bf16.h>

// ---------------------------------------------------------------------------
// ACT loss head for MI455X (gfx1250).
// Roofline: one pass over 210MB of fp32 logits -> ~9us @ 23.3 TB/s. Round-1
// wave-per-token shuffles (ds_bpermute) made kernel A instruction-bound
// (~150 instr/token); this version is thread-per-token with b128 loads and
// register-serial reductions (~7 instr/token) -> HBM-bound again.
// CDNA5-specific paths:
//   * global_load_async_to_lds_b128 + s_wait_asynccnt (ASYNCcnt) staging of
//     the per-row prediction grid into LDS (kernel B)
//   * wave32 codegen throughout (gfx1250 is wave32-only)
// No WMMA on purpose: the reference contains no matmul.
// ---------------------------------------------------------------------------

#define GRID_W        40
#define SEQ           1600          // GRID_W * GRID_W
#define VOCAB         32
#define PATH_TOK      6
#define IGNORE_ID     (-100)
#define SP_W          10.0f
#define CONN_W        5.0f
#define TOK_PER_BLOCK 320           // 10 waves, 1 token per thread
#define BLOCKS_PER_ROW (SEQ / TOK_PER_BLOCK)   // 5
#define BIG_LABEL     (1 << 30)

// ---------------------------------------------------------------------------
// Kernel A: per-token argmax + cross-entropy, one THREAD per token.
// 8 x global_load_b128 per thread; max/argmax/sum-exp fully in registers.
// ---------------------------------------------------------------------------
__global__ __launch_bounds__(TOK_PER_BLOCK)
void ce_argmax_kernel(const float* __restrict__ logits,
                      const int*   __restrict__ labels,
                      unsigned char* __restrict__ pred,
                      float* __restrict__ ce_part,
                      int*   __restrict__ cnt_part)
{
    const int t     = threadIdx.x;
    const int b     = blockIdx.x / BLOCKS_PER_ROW;
    const int chunk = blockIdx.x % BLOCKS_PER_ROW;
    const int s     = chunk * TOK_PER_BLOCK + t;
    const long tok  = (long)b * SEQ + s;

    // 32 logits for this token via 8 x 16B loads
    const float4* __restrict__ p = (const float4*)(logits + tok * VOCAB);
    float xs[VOCAB];
#pragma unroll
    for (int k = 0; k < VOCAB / 4; ++k) {
        float4 q = p[k];
        xs[4 * k + 0] = q.x; xs[4 * k + 1] = q.y;
        xs[4 * k + 2] = q.z; xs[4 * k + 3] = q.w;
    }

    // max (serial chain in registers)
    float m = xs[0];
#pragma unroll
    for (int i = 1; i < VOCAB; ++i) m = fmaxf(m, xs[i]);

    // first-occurrence argmax: scan descending so lowest index wins
    int midx = 0;
#pragma unroll
    for (int i = VOCAB - 1; i >= 0; --i) if (xs[i] == m) midx = i;

    // sum of exp(x - m)
    float ssum = 0.0f;
#pragma unroll
    for (int i = 0; i < VOCAB; ++i) ssum += __expf(xs[i] - m);
    float logZ = m + __logf(ssum);

    int   label = labels[tok];
    int   maskv = (label != IGNORE_ID) ? 1 : 0;
    int   srcl  = (label >= 0 && label < VOCAB) ? label : 0;
    float xl    = logits[tok * VOCAB + srcl];     // L0 hit: line just loaded
    float ce    = maskv ? (logZ - xl) : 0.0f;
    int   corr  = (maskv && (midx == label)) ? 1 : 0;

    pred[tok] = (unsigned char)midx;

    // fixed-order block reduction (deterministic)
    __shared__ float s_redf[TOK_PER_BLOCK];
    __shared__ int   s_redi[TOK_PER_BLOCK];
    s_redf[t] = ce;
    s_redi[t] = (corr << 16) | maskv;
    __syncthreads();
#pragma unroll
    for (int str = TOK_PER_BLOCK / 2; str >= 5; str >>= 1) {   // 160..5
        if (t < str) { s_redf[t] += s_redf[t + str]; s_redi[t] += s_redi[t + str]; }
        __syncthreads();
    }
    if (t == 0) {
        float cs = 0.0f; int cc = 0;
#pragma unroll
        for (int k = 0; k < 5; ++k) { cs += s_redf[k]; cc += s_redi[k]; }
        ce_part[blockIdx.x]  = cs;
        cnt_part[blockIdx.x] = cc;
    }
}

// ---------------------------------------------------------------------------
// Kernel B: one block per batch row. Async-copies the 1600-byte pred row into
// LDS (CDNA5 ASYNCcnt path), then spatial penalty + LDS connected components
// + CE/BCE fold-in. All reductions fixed-order.
// ---------------------------------------------------------------------------
__global__ __launch_bounds__(256)
void row_kernel(const unsigned char* __restrict__ pred,
                const float* __restrict__ ce_part,
                const int*   __restrict__ cnt_part,
                const float* __restrict__ q_halt,
                float* __restrict__ row_total,
                int B)
{
    const int b = blockIdx.x;
    const int t = threadIdx.x;

    __shared__ __align__(16) unsigned char s_pred[SEQ];   // 1600 B
    __shared__ int   s_lab[SEQ];                          // 6400 B
    __shared__ float s_redf[256];
    __shared__ int   s_redi[256];
    __shared__ int   s_changed;

    // --- CDNA5 async global -> LDS copy of the row (100 x 16B), ASYNCcnt ---
    if (t < SEQ / 16) {
        unsigned int       lds_addr = (unsigned int)(size_t)(&s_pred[t * 16]);
        unsigned long long gaddr    =
            (unsigned long long)(size_t)(pred + (size_t)b * SEQ + (size_t)t * 16);
        asm volatile("global_load_async_to_lds_b128 %0, %1, off"
                     :: "v"(lds_addr), "v"(gaddr) : "memory");
    }
    asm volatile("s_wait_asynccnt 0x0" ::: "memory");
    __syncthreads();

    // --- spatial penalty: for each path cell, Manhattan jump to next path ---
    float myp = 0.0f;
    for (int i = t; i < SEQ; i += 256) {
        if (s_pred[i] == PATH_TOK) {
            int j = i + 1;
            while (j < SEQ && s_pred[j] != PATH_TOK) ++j;
            if (j < SEQ) {
                int r1 = i / GRID_W, c1 = i % GRID_W;
                int r2 = j / GRID_W, c2 = j % GRID_W;
                int dr = r1 > r2 ? r1 - r2 : r2 - r1;
                int dc = c1 > c2 ? c1 - c2 : c2 - c1;
                int d  = dr + dc;
                if (d > 1) myp += (float)(d - 1) * SP_W;
            }
        }
    }
    s_redf[t] = myp;

    // --- connectivity: in-LDS min-label propagation to the unique fixpoint --
    for (int i = t; i < SEQ; i += 256)
        s_lab[i] = (s_pred[i] == PATH_TOK) ? i : BIG_LABEL;

    for (;;) {
        __syncthreads();
        if (t == 0) s_changed = 0;
        __syncthreads();
        int ch = 0;
        for (int i = t; i < SEQ; i += 256) {
            if (s_pred[i] == PATH_TOK) {
                int r = i / GRID_W, c = i % GRID_W;
                int m = s_lab[i];
                if (r > 0)          m = min(m, s_lab[i - GRID_W]);
                if (r < GRID_W - 1) m = min(m, s_lab[i + GRID_W]);
                if (c > 0)          m = min(m, s_lab[i - 1]);
                if (c < GRID_W - 1) m = min(m, s_lab[i + 1]);
                if (m < s_lab[i]) { s_lab[i] = m; ch = 1; }
            }
        }
        if (ch) s_changed = 1;
        __syncthreads();
        if (s_changed == 0) break;
    }

    // component roots: path cell whose label equals its own index
    int myc = 0;
    for (int i = t; i < SEQ; i += 256)
        if (s_pred[i] == PATH_TOK && s_lab[i] == i) ++myc;
    s_redi[t] = myc;
    __syncthreads();

    if (t == 0) {
        float sp = 0.0f; int comp = 0;
        for (int k = 0; k < 256; ++k) { sp += s_redf[k]; comp += s_redi[k]; }
        float conn = (comp > 1) ? (float)(comp - 1) * CONN_W : 0.0f;

        float cs = 0.0f; int cc = 0;
#pragma unroll
        for (int k = 0; k < BLOCKS_PER_ROW; ++k) {
            cs += ce_part[b * BLOCKS_PER_ROW + k];
            cc += cnt_part[b * BLOCKS_PER_ROW + k];
        }
        int maskc = cc & 0xFFFF;
        int corrc = cc >> 16;
        float divisor = (float)(maskc > 1 ? maskc : 1);
        float lm = cs / divisor;

        int   seq_ok = (corrc == maskc) ? 1 : 0;
        float xq  = q_halt[b];
        float tq  = (float)seq_ok;
        float bce = fmaxf(xq, 0.0f) - xq * tq + log1pf(expf(-fabsf(xq)));

        row_total[b] = lm + 0.5f * bce + (sp + conn) / (float)B;
    }
}

// ---------------------------------------------------------------------------
// Kernel C: fixed-order reduction of per-row totals -> scalar loss.
// ---------------------------------------------------------------------------
__global__ __launch_bounds__(256)
void final_kernel(const float* __restrict__ row_total,
                  float* __restrict__ out, int B)
{
    __shared__ float s[256];
    float a = 0.0f;
    for (int i = threadIdx.x; i < B; i += 256) a += row_total[i];
    s[threadIdx.x] = a;
    __syncthreads();
    if (threadIdx.x == 0) {
        float tot = 0.0f;
        for (int k = 0; k < 256; ++k) tot += s[k];
        out[0] = tot;
    }
}

// ---------------------------------------------------------------------------
// Launch. Inputs (setup_inputs order):
//   d_in[0] logits  [B,1600,32] f32   d_in[1] labels [B,1600] int
//   d_in[2] q_halt  [B] f32           d_in[3] halted (unused)
//   d_in[4] steps (unused)
// Workspace layout (B=1024 -> ~1.7 MB):
//   ce_part f32[B*5] | cnt_part i32[B*5] | pred u8[B*1600] | row f32[B]
// ---------------------------------------------------------------------------
extern "C" void kernel_launch(void* const* d_in, const int* in_sizes, int n_in,
                              void* d_out, int out_size, void* d_ws, size_t ws_size,
                              hipStream_t stream)
{
    (void)n_in; (void)out_size; (void)ws_size;

    const float* logits = (const float*)d_in[0];
    const int*   labels = (const int*)d_in[1];
    const float* q_halt = (const float*)d_in[2];
    const int B = in_sizes[2];                    // q_halt_logits is [B]

    char* ws = (char*)d_ws;
    float*         ce_part   = (float*)ws;
    int*           cnt_part  = (int*)(ws + (size_t)B * BLOCKS_PER_ROW * 4);
    unsigned char* pred      = (unsigned char*)(ws + (size_t)B * BLOCKS_PER_ROW * 8);
    float*         row_total = (float*)(ws + (size_t)B * BLOCKS_PER_ROW * 8 + (size_t)B * SEQ);

    ce_argmax_kernel<<<B * BLOCKS_PER_ROW, TOK_PER_BLOCK, 0, stream>>>(
        logits, labels, pred, ce_part, cnt_part);
    row_kernel<<<B, 256, 0, stream>>>(
        pred, ce_part, cnt_part, q_halt, row_total, B);
    final_kernel<<<1, 256, 0, stream>>>(row_total, (float*)d_out, B);
}